// Gryoautomorphism_32323923870229
// MI455X (gfx1250) — compile-verified
//
#include <hip/hip_runtime.h>

// ---------------------------------------------------------------------------
// MI455X (gfx1250).
//   Kernel A (matfn_kernel): per-channel matrix functions via matmul-only
//     algorithms (Horner expm, Newton-Schulz sqrt/invsqrt/inverse), every
//     64x64 matmul done with V_WMMA_F32_16X16X4_F32 (full f32 precision).
//   Kernel B (apply_kernel): streaming out[b,c] = F[c] @ x[b,c] @ invF[c].
// All LDS buffers are addressed as S[bufIdx][...] through a reference to the
// __shared__ array so clang's InferAddressSpaces keeps every access in
// addrspace(3): 32-bit ds ops with immediate offsets, no flat null-checks.
// ---------------------------------------------------------------------------

#define LS 68              // LDS row stride in floats (pad 64->68: conflict-free, 16B rows)
#define MATE (64 * LS)     // floats per 64x64 LDS matrix buffer
#define NTHREADS 512       // 16 waves per workgroup (wave32)

typedef __attribute__((ext_vector_type(2))) float v2f;
typedef __attribute__((ext_vector_type(8))) float v8f;

// ---- one 16x16 output tile of a 64x64 product, K-loop of 16 WMMAs ---------
template <int NB>
__device__ __forceinline__ v8f tile_mm16(const float (&S)[NB][MATE], int ai, int bi,
                                         int m0, int n0, int mr, int kh) {
  v8f acc = {};
  // lanes 0-15: K = k0+0,k0+1 ; lanes 16-31: K = k0+2,k0+3
  const int abase = ai * MATE + (m0 + mr) * LS + 2 * kh;        // A row, this lane's K pair
  const int bbase = bi * MATE + (2 * kh) * LS + n0 + mr;        // B col, this lane's K pair
  const float* Sf = &S[0][0];
#pragma unroll
  for (int k0 = 0; k0 < 64; k0 += 4) {
    v2f a = *(const v2f*)(Sf + abase + k0);                      // ds_load_b64, imm offset
    v2f b;
    b.x = Sf[bbase + k0 * LS];                                   // ds_load_2addr pair
    b.y = Sf[bbase + k0 * LS + LS];
    acc = __builtin_amdgcn_wmma_f32_16x16x4_f32(
        /*neg_a=*/false, a, /*neg_b=*/false, b,
        /*c_mod=*/(short)0, acc, /*reuse_a=*/false, /*reuse_b=*/false);
  }
  return acc;
}

// ---- S[ci] = s*(S[ai]@S[bi]) + dI*I ; 16 waves, one 16x16 tile each -------
template <int NB>
__device__ __forceinline__ void wg_mm(float (&S)[NB][MATE], int ci, int ai, int bi,
                                      float s, float dI) {
  __syncthreads();                        // operand writes from all waves visible
  const int w = threadIdx.x >> 5, lane = threadIdx.x & 31;
  const int m0 = (w >> 2) << 4, n0 = (w & 3) << 4;
  const int mr = lane & 15, kh = lane >> 4;
  v8f acc = tile_mm16(S, ai, bi, m0, n0, mr, kh);
#pragma unroll
  for (int v = 0; v < 8; ++v) {
    const int row = m0 + v + 8 * kh, col = n0 + mr;   // C/D VGPR layout
    S[ci][row * LS + col] = s * acc[v] + ((row == col) ? dI : 0.0f);
  }
  __syncthreads();                        // result complete before anyone reads it
}

// ---------------- elementwise helpers (512 threads over 4096 elems) --------
template <int NB>
__device__ __forceinline__ void wg_sym_scaled(float (&S)[NB][MATE], int di,
                                              const float* __restrict__ g, float s) {
  for (int e = threadIdx.x; e < 4096; e += NTHREADS) {
    const int r = e >> 6, c = e & 63;
    S[di][r * LS + c] = s * (g[r * 64 + c] + g[c * 64 + r]);
  }
  __syncthreads();
}
template <int NB>
__device__ __forceinline__ void wg_axpyI(float (&S)[NB][MATE], int di, int si,
                                         float s, float d) {
  for (int e = threadIdx.x; e < 4096; e += NTHREADS) {
    const int r = e >> 6, c = e & 63;
    S[di][r * LS + c] = s * S[si][r * LS + c] + ((r == c) ? d : 0.0f);
  }
  __syncthreads();
}
template <int NB>
__device__ __forceinline__ void wg_scale_from(float (&S)[NB][MATE], int di, int si, float s) {
  for (int e = threadIdx.x; e < 4096; e += NTHREADS) {
    const int r = e >> 6, c = e & 63;
    S[di][r * LS + c] = s * S[si][r * LS + c];
  }
  __syncthreads();
}
template <int NB>
__device__ __forceinline__ void wg_setI(float (&S)[NB][MATE], int di) {
  for (int e = threadIdx.x; e < 4096; e += NTHREADS) {
    const int r = e >> 6, c = e & 63;
    S[di][r * LS + c] = (r == c) ? 1.0f : 0.0f;
  }
  __syncthreads();
}
template <int NB>
__device__ __forceinline__ void wg_transpose_scale(float (&S)[NB][MATE], int di, int si, float s) {
  for (int e = threadIdx.x; e < 4096; e += NTHREADS) {
    const int r = e >> 6, c = e & 63;
    S[di][r * LS + c] = s * S[si][c * LS + r];
  }
  __syncthreads();
}
template <int NB>
__device__ __forceinline__ void wg_copy(float (&S)[NB][MATE], int di, int si) {
  for (int e = threadIdx.x; e < 4096; e += NTHREADS) {
    const int r = e >> 6, c = e & 63;
    S[di][r * LS + c] = S[si][r * LS + c];
  }
  __syncthreads();
}
template <int NB>
__device__ __forceinline__ void wg_store_global(float* __restrict__ g,
                                                const float (&S)[NB][MATE], int si) {
  for (int e = threadIdx.x; e < 4096; e += NTHREADS) {
    const int r = e >> 6, c = e & 63;
    g[r * 64 + c] = S[si][r * LS + c];
  }
  __syncthreads();
}
template <int NB>
__device__ __forceinline__ float wg_frob2(const float (&S)[NB][MATE], int si, float& red) {
  __syncthreads();
  if (threadIdx.x == 0) red = 0.0f;
  __syncthreads();
  float p = 0.0f;
  for (int e = threadIdx.x; e < 4096; e += NTHREADS) {
    const int r = e >> 6, c = e & 63;
    const float v = S[si][r * LS + c];
    p += v * v;
  }
  atomicAdd(&red, p);
  __syncthreads();
  return red;
}

// ---- expm(sym(g)) via scaling-and-squaring (2^-4 scale, Taylor deg 8) -----
template <int NB>
__device__ __forceinline__ void dev_expm(const float* __restrict__ g,
                                         float (&S)[NB][MATE],
                                         int out, int bM, int b1, int b2) {
  wg_sym_scaled(S, bM, g, 0.5f / 16.0f);          // M = sym(g)/16
  wg_axpyI(S, b1, bM, 1.0f / 8.0f, 1.0f);         // R = I + M/8
  int R = b1, Rn = b2;
#pragma unroll 1
  for (int k = 7; k >= 1; --k) {                  // R = I + (M@R)/k
    wg_mm(S, Rn, bM, R, 1.0f / (float)k, 1.0f);
    int t = R; R = Rn; Rn = t;
  }
#pragma unroll 1
  for (int i = 0; i < 4; ++i) {                   // square 4 times
    wg_mm(S, Rn, R, R, 1.0f, 0.0f);
    int t = R; R = Rn; Rn = t;
  }
  if (out != R) wg_copy(S, out, R);
}

// ---- coupled Newton-Schulz: Y->sqrt(A), Z->invsqrt(A). S[ia] preserved. ---
template <int NB>
__device__ __forceinline__ void dev_ns_sqrt(float (&S)[NB][MATE], int ia,
                                            int b1, int b2, int b3, int bT,
                                            float& red, int outSqrt, int outInv,
                                            int iters) {
  const float n2 = wg_frob2(S, ia, red);
  const float nrm = sqrtf(n2);
  wg_scale_from(S, b1, ia, 1.0f / nrm);           // Y0 = A/||A||_F
  wg_setI(S, b2);                                 // Z0 = I
  int Y = b1, Z = b2, sp = b3;
#pragma unroll 1
  for (int it = 0; it < iters; ++it) {
    wg_mm(S, bT, Z, Y, -0.5f, 1.5f);              // T = 1.5 I - 0.5 Z Y
    wg_mm(S, sp, Y, bT, 1.0f, 0.0f);              // Ynew = Y T
    wg_mm(S, Y, bT, Z, 1.0f, 0.0f);               // Znew = T Z (old Y dead)
    const int nY = sp, nZ = Y, nS = Z;
    Y = nY; Z = nZ; sp = nS;
  }
  if (outSqrt >= 0) wg_scale_from(S, outSqrt, Y, sqrtf(nrm));
  if (outInv >= 0)  wg_scale_from(S, outInv, Z, rsqrtf(nrm));
}

// ---- Newton-Schulz inverse: X <- X(2I - F X), X0 = F^T/||F||_F^2 ----------
template <int NB>
__device__ __forceinline__ void dev_ns_inv(float (&S)[NB][MATE], int iF,
                                           int b1, int b2, int bT, float& red,
                                           float* __restrict__ gout) {
  const float n2 = wg_frob2(S, iF, red);
  wg_transpose_scale(S, b1, iF, 1.0f / n2);
  int X = b1, sp = b2;
#pragma unroll 1
  for (int it = 0; it < 28; ++it) {
    wg_mm(S, bT, iF, X, -1.0f, 2.0f);             // T = 2I - F X
    wg_mm(S, sp, X, bT, 1.0f, 0.0f);              // X = X T
    const int t = X; X = sp; sp = t;
  }
  wg_store_global(gout, S, X);
}

// ---------------------------------------------------------------------------
// Kernel A: one workgroup per channel. Builds F[c], invF[c] into workspace.
// ---------------------------------------------------------------------------
__global__ __launch_bounds__(NTHREADS) void matfn_kernel(
    const float* __restrict__ Pp, const float* __restrict__ Qp,
    float* __restrict__ dF, float* __restrict__ dG) {
  __shared__ float S[7][MATE];                    // ~119 KB of the WGP's 320 KB LDS
  __shared__ float red;
  const int c = blockIdx.x;
  const float* gP = Pp + (size_t)c * 4096;
  const float* gQ = Qp + (size_t)c * 4096;

  // exp(sym(P)) -> S0 ; sqrtP -> S5
  dev_expm(gP, S, 0, 1, 2, 3);
  dev_ns_sqrt(S, 0, 1, 2, 3, 4, red, /*sqrt*/5, /*inv*/-1, 14);

  // exp(sym(Q)) -> S0 (kept) ; sqrtQ -> S6
  dev_expm(gQ, S, 0, 1, 2, 3);
  dev_ns_sqrt(S, 0, 1, 2, 3, 4, red, /*sqrt*/6, /*inv*/-1, 14);

  // M = sqrtP @ expQ @ sqrtP -> S2
  wg_mm(S, 1, 5, 0, 1.0f, 0.0f);
  wg_mm(S, 2, 1, 5, 1.0f, 0.0f);

  // invsqrt(M) -> S2 (A aliasing dst OK: A only read at NS init)
  dev_ns_sqrt(S, 2, 0, 1, 3, 4, red, /*sqrt*/-1, /*inv*/2, 18);

  // F = invsqrtM @ sqrtP @ sqrtQ -> S1 ; store
  wg_mm(S, 0, 2, 5, 1.0f, 0.0f);
  wg_mm(S, 1, 0, 6, 1.0f, 0.0f);
  wg_store_global(dF + (size_t)c * 4096, S, 1);

  // invF via Newton-Schulz ; store
  dev_ns_inv(S, 1, 2, 3, 4, red, dG + (size_t)c * 4096);
}

// ---------------------------------------------------------------------------
// Kernel B: one workgroup per (b,c). out = F[c] @ x[b,c] @ invF[c].
// ---------------------------------------------------------------------------
template <int NB>
__device__ __forceinline__ void load64(float (&S)[NB][MATE], int di,
                                       const float* __restrict__ g) {
  const int t = threadIdx.x;               // 512 threads x 8 floats (2x float4)
  const int r = t >> 3, q = t & 7;
  const float4* gs = (const float4*)(g + r * 64 + q * 8);
  float4 v0 = gs[0];
  float4 v1 = gs[1];
  float4* ds = (float4*)(&S[di][r * LS + q * 8]);  // 16B aligned (LS*4 = 272)
  ds[0] = v0;
  ds[1] = v1;
}

__global__ __launch_bounds__(NTHREADS) void apply_kernel(
    const float* __restrict__ x, const float* __restrict__ dF,
    const float* __restrict__ dG, float* __restrict__ out) {
  __shared__ float S[4][MATE];             // X, F, G(invF), T
  const int idx = blockIdx.x;
  const int c = idx >> 6;                  // consecutive blocks share c -> F/invF stay hot
  const int b = idx & 63;
  const float* gx = x + ((size_t)b * 128 + c) * 4096;
  float* go = out + ((size_t)b * 128 + c) * 4096;

  load64(S, 0, gx);
  load64(S, 1, dF + (size_t)c * 4096);
  load64(S, 2, dG + (size_t)c * 4096);

  // T = F @ x   (wg_mm provides the barriers)
  wg_mm(S, 3, 1, 0, 1.0f, 0.0f);

  // out = T @ invF, stored straight from the WMMA accumulators
  const int w = threadIdx.x >> 5, lane = threadIdx.x & 31;
  const int m0 = (w >> 2) << 4, n0 = (w & 3) << 4;
  const int mr = lane & 15, kh = lane >> 4;
  v8f acc = tile_mm16(S, 3, 2, m0, n0, mr, kh);
#pragma unroll
  for (int v = 0; v < 8; ++v) {
    const int row = m0 + v + 8 * kh, col = n0 + mr;
    go[row * 64 + col] = acc[v];
  }
}

// ---------------------------------------------------------------------------
extern "C" void kernel_launch(void* const* d_in, const int* in_sizes, int n_in,
                              void* d_out, int out_size, void* d_ws, size_t ws_size,
                              hipStream_t stream) {
  (void)in_sizes; (void)n_in; (void)out_size; (void)ws_size;
  const float* x = (const float*)d_in[0];        // [64,128,64,64]
  const float* P = (const float*)d_in[1];        // [128,64,64]
  const float* Q = (const float*)d_in[2];        // [128,64,64]
  float* out = (float*)d_out;                    // [64,128,64,64]
  float* dF = (float*)d_ws;                      // F:    128*4096 floats (2 MiB)
  float* dG = dF + 128 * 4096;                   // invF: 128*4096 floats (2 MiB)

  matfn_kernel<<<128, NTHREADS, 0, stream>>>(P, Q, dF, dG);
  apply_kernel<<<64 * 128, NTHREADS, 0, stream>>>(x, dF, dG, out);
}